// SpatialEncoder_18915035972036
// MI455X (gfx1250) — compile-verified
//
#include <hip/hip_runtime.h>

typedef float v4f __attribute__((ext_vector_type(4)));

namespace {
constexpr int kGrid   = 8;     // region grid
constexpr int kNBits  = 64;
constexpr int kB      = 16;
constexpr int kH      = 512;
constexpr int kW      = 512;
constexpr int kRowsPerBlock = 16;           // strip height (aligned inside 64-row regions)
constexpr int kRowsPerIter  = 2;            // 2 rows * 512 f32 = 4 KB per stage
constexpr int kThreads      = 256;          // 256 lanes * 16B = 4 KB = one stage
constexpr int kIters        = kRowsPerBlock / kRowsPerIter;  // 8
constexpr int kStrips       = kH / kRowsPerBlock;            // 32
}

__global__ __launch_bounds__(kThreads)
void spatial_encoder_kernel(const float* __restrict__ M,
                            const int*   __restrict__ bits,
                            const float* __restrict__ W_inj,
                            const float* __restrict__ b_inj,
                            float*       __restrict__ out)
{
    __shared__ float sLUT[3 * kNBits];              // per-(channel,region) scale
    __shared__ float mbuf[2][kThreads * 4];         // double-buffered M stage (2 x 4 KB)

    const int t     = threadIdx.x;
    const int strip = blockIdx.x & (kStrips - 1);
    const int b     = blockIdx.x / kStrips;

    // Phase 0: build s[c][r] = (pol[b][r] * W_inj[c][r] + b_inj[c]) * 0.5
    if (t < 3 * kNBits) {
        const int c = t >> 6;
        const int r = t & 63;
        const float pol = (float)bits[b * kNBits + r] * 2.0f - 1.0f;
        sLUT[t] = (pol * W_inj[c * kNBits + r] + b_inj[c]) * 0.5f;
    }
    __syncthreads();

    // Thread-invariant geometry: lane t covers row (t>>7) of each 2-row stage,
    // float4 chunk (t&127) within the row.
    const int rowInStage = t >> 7;                  // 0..1
    const int chunk      = t & 127;                 // 0..127
    const int w0         = chunk << 2;              // column of first element
    const int wreg       = w0 >> 6;                 // region column
    const int hreg       = (strip * kRowsPerBlock) >> 6;   // region row (strip-invariant)
    const int reg        = hreg * kGrid + wreg;

    const float s0 = sLUT[reg];
    const float s1 = sLUT[kNBits + reg];
    const float s2 = sLUT[2 * kNBits + reg];

    const long long row0 = (long long)strip * kRowsPerBlock;
    const float* mBase = M   + ((long long)b * kH + row0) * kW;
    float*       oBase = out + (((long long)b * 3) * kH + row0) * kW;
    const float* gsrc0 = mBase + (long long)rowInStage * kW + w0;

    // Per-lane LDS byte offsets of this lane's 16B staging slot (addrspace(3) offset
    // == low 32 bits of the flat address per the LDS aperture mapping).
    unsigned ldsOff0 = (unsigned)(unsigned long long)(const void*)&mbuf[0][t * 4];
    unsigned ldsOff1 = (unsigned)(unsigned long long)(const void*)&mbuf[1][t * 4];

    // Stage 0: kick off the first async global->LDS copy (ASYNCcnt path).
    {
        unsigned long long ga = (unsigned long long)gsrc0;
        asm volatile("global_load_async_to_lds_b128 %0, %1, off"
                     :: "v"(ldsOff0), "v"(ga) : "memory");
    }

#pragma unroll
    for (int i = 0; i < kIters; ++i) {
        // Prefetch stage i+1 into the other buffer, then retire stage i.
        if (i + 1 < kIters) {
            unsigned long long ga =
                (unsigned long long)(gsrc0 + (long long)(i + 1) * kRowsPerIter * kW);
            unsigned dst = ((i + 1) & 1) ? ldsOff1 : ldsOff0;
            asm volatile("global_load_async_to_lds_b128 %0, %1, off"
                         :: "v"(dst), "v"(ga) : "memory");
            // Two async loads in flight; they complete in order, so cnt<=1
            // guarantees stage i's data is in LDS.
            asm volatile("s_wait_asynccnt 0x1" ::: "memory");
        } else {
            asm volatile("s_wait_asynccnt 0x0" ::: "memory");
        }

        // Each lane reads back exactly the 16B its own wave's async op wrote.
        v4f m4 = *(const v4f*)&mbuf[i & 1][t * 4];

        const long long rowOff =
            (long long)(i * kRowsPerIter + rowInStage) * kW + w0;

        v4f o0 = m4 * s0;
        v4f o1 = m4 * s1;
        v4f o2 = m4 * s2;

        // Streaming output: written once, never re-read -> non-temporal b128 stores.
        __builtin_nontemporal_store(o0, (v4f*)(oBase + rowOff));
        __builtin_nontemporal_store(o1, (v4f*)(oBase + (long long)kH * kW + rowOff));
        __builtin_nontemporal_store(o2, (v4f*)(oBase + 2LL * kH * kW + rowOff));
    }
}

extern "C" void kernel_launch(void* const* d_in, const int* in_sizes, int n_in,
                              void* d_out, int out_size, void* d_ws, size_t ws_size,
                              hipStream_t stream) {
    // setup_inputs order: img (unused beyond shape), M, bits, W_inj, b_inj
    const float* M     = (const float*)d_in[1];
    const int*   bits  = (const int*)d_in[2];
    const float* W_inj = (const float*)d_in[3];
    const float* b_inj = (const float*)d_in[4];
    float*       out   = (float*)d_out;

    dim3 grid(kB * kStrips);   // 512 blocks
    dim3 block(kThreads);      // 256 threads (8 wave32)
    spatial_encoder_kernel<<<grid, block, 0, stream>>>(M, bits, W_inj, b_inj, out);
}